// SpectralBasedGraphConvLayer_37280316129403
// MI455X (gfx1250) — compile-verified
//
#include <hip/hip_runtime.h>

// Spectral graph conv on MI455X (gfx1250):
//   y      = U^T @ x                  [8192 x 32]
//   scaled = y * filters^T            (fused into GEMM1 epilogue)
//   out    = U @ scaled               [8192 x 32]
//
// Bandwidth-bound: U (256 MB) streamed exactly once per GEMM -> ~512 MB HBM
// -> ~22 us floor at 23.3 TB/s. Compute via V_WMMA_F32_16X16X4_F32.

typedef __attribute__((ext_vector_type(2))) float v2f;
typedef __attribute__((ext_vector_type(8))) float v8f;

#define N_DIM 8192
#define F_DIM 32
#define WAVES_PER_BLOCK 8
#define K_PER_WAVE (N_DIM / WAVES_PER_BLOCK) // 1024

// ---------------------------------------------------------------------------
// GEMM1: ysc[i, f] = (sum_k U[k, i] * x[k, f]) * filters[f, i]
// One block per 16-row strip (i0). 8 waves K-split; LDS reduce (deterministic).
// A fragment (16x4 f32, ISA 7.12.2): lane<16 -> M=lane, K={0,1};
//                                    lane>=16 -> M=lane-16, K={2,3}.
// B fragment (4x16 f32, mirrored):   lane<16 -> N=lane, K={0,1};
//                                    lane>=16 -> N=lane-16, K={2,3}.
// C/D (16x16 f32): vgpr v, lane<16 -> M=v, N=lane; lane>=16 -> M=v+8.
// ---------------------------------------------------------------------------
__global__ __launch_bounds__(256) void gemm_ut_x_scale(
    const float* __restrict__ U, const float* __restrict__ x,
    const float* __restrict__ filters, float* __restrict__ ysc) {
  __shared__ float red[WAVES_PER_BLOCK * 512]; // 16 KB

  const unsigned lane = threadIdx.x & 31u;
  const unsigned wave = threadIdx.x >> 5;
  const unsigned i0 = blockIdx.x * 16u;
  const unsigned m = lane & 15u;        // row (A) / col (B) within tile
  const unsigned kb = (lane >> 4) << 1; // 0 or 2 (K sub-pair per half-wave)
  const unsigned k0 = wave * K_PER_WAVE;

  // A: U^T[i0+m, k] = U[k*N + i0 + m]  (lanes 0-15 contiguous 64B per row)
  const float* Abase = U + (size_t)(k0 + kb) * N_DIM + i0 + m;
  // B: x[k, f0+n], n = lane&15, two F-tiles at f0=0 and f0=16
  const float* Bbase = x + (size_t)(k0 + kb) * F_DIM + m;

  v8f acc0 = {};
  v8f acc1 = {};

#pragma unroll 8
  for (unsigned ks = 0; ks < K_PER_WAVE; ks += 4) {
    v2f a, b0, b1;
    a.x = Abase[(size_t)ks * N_DIM];
    a.y = Abase[(size_t)(ks + 1) * N_DIM];
    b0.x = Bbase[(size_t)ks * F_DIM];
    b0.y = Bbase[(size_t)(ks + 1) * F_DIM];
    b1.x = Bbase[(size_t)ks * F_DIM + 16];
    b1.y = Bbase[(size_t)(ks + 1) * F_DIM + 16];
    acc0 = __builtin_amdgcn_wmma_f32_16x16x4_f32(false, a, false, b0,
                                                 (short)0, acc0, false, false);
    acc1 = __builtin_amdgcn_wmma_f32_16x16x4_f32(false, a, false, b1,
                                                 (short)0, acc1, false, false);
  }

  // Scatter accumulators to LDS in logical (m, f) order.
  const unsigned mbase = (lane >> 4) << 3; // 0 or 8
  const unsigned nn = lane & 15u;
#pragma unroll
  for (unsigned v = 0; v < 8; ++v) {
    red[wave * 512 + (mbase + v) * 32 + nn] = acc0[v];
    red[wave * 512 + (mbase + v) * 32 + 16 + nn] = acc1[v];
  }
  __syncthreads();

  // Deterministic cross-wave reduce + fused spectral filter scale.
#pragma unroll
  for (unsigned e = threadIdx.x; e < 512; e += 256) {
    float s = 0.f;
#pragma unroll
    for (unsigned w = 0; w < WAVES_PER_BLOCK; ++w) s += red[w * 512 + e];
    const unsigned mm = e >> 5, cc = e & 31u; // (row-in-strip, feature)
    s *= filters[(size_t)cc * N_DIM + i0 + mm];
    ysc[(size_t)i0 * F_DIM + e] = s; // coalesced
  }
}

// ---------------------------------------------------------------------------
// GEMM2: out[i, f] = sum_k U[i, k] * ysc[k, f]
// A now row-major per lane -> packed float2 loads; cacheline reuse across the
// unrolled K steps keeps HBM traffic at exactly one pass over U.
// ---------------------------------------------------------------------------
__global__ __launch_bounds__(256) void gemm_u_y(
    const float* __restrict__ U, const float* __restrict__ ysc,
    float* __restrict__ out) {
  __shared__ float red[WAVES_PER_BLOCK * 512];

  const unsigned lane = threadIdx.x & 31u;
  const unsigned wave = threadIdx.x >> 5;
  const unsigned i0 = blockIdx.x * 16u;
  const unsigned m = lane & 15u;
  const unsigned kb = (lane >> 4) << 1;
  const unsigned k0 = wave * K_PER_WAVE;

  // A: U[i0+m, k0+ks+kb .. +1]  (8B-aligned: kb even, ks%4==0, k0%1024==0)
  const float* Arow = U + (size_t)(i0 + m) * N_DIM + k0 + kb;
  const float* Bbase = ysc + (size_t)(k0 + kb) * F_DIM + m;

  v8f acc0 = {};
  v8f acc1 = {};

#pragma unroll 8
  for (unsigned ks = 0; ks < K_PER_WAVE; ks += 4) {
    v2f a = *(const v2f*)(Arow + ks);
    v2f b0, b1;
    b0.x = Bbase[(size_t)ks * F_DIM];
    b0.y = Bbase[(size_t)(ks + 1) * F_DIM];
    b1.x = Bbase[(size_t)ks * F_DIM + 16];
    b1.y = Bbase[(size_t)(ks + 1) * F_DIM + 16];
    acc0 = __builtin_amdgcn_wmma_f32_16x16x4_f32(false, a, false, b0,
                                                 (short)0, acc0, false, false);
    acc1 = __builtin_amdgcn_wmma_f32_16x16x4_f32(false, a, false, b1,
                                                 (short)0, acc1, false, false);
  }

  const unsigned mbase = (lane >> 4) << 3;
  const unsigned nn = lane & 15u;
#pragma unroll
  for (unsigned v = 0; v < 8; ++v) {
    red[wave * 512 + (mbase + v) * 32 + nn] = acc0[v];
    red[wave * 512 + (mbase + v) * 32 + 16 + nn] = acc1[v];
  }
  __syncthreads();

#pragma unroll
  for (unsigned e = threadIdx.x; e < 512; e += 256) {
    float s = 0.f;
#pragma unroll
    for (unsigned w = 0; w < WAVES_PER_BLOCK; ++w) s += red[w * 512 + e];
    out[(size_t)i0 * F_DIM + e] = s;
  }
}

extern "C" void kernel_launch(void* const* d_in, const int* in_sizes, int n_in,
                              void* d_out, int out_size, void* d_ws,
                              size_t ws_size, hipStream_t stream) {
  const float* x = (const float*)d_in[0];       // [8192, 32]
  const float* U = (const float*)d_in[1];       // [8192, 8192]
  const float* filters = (const float*)d_in[2]; // [32, 8192]
  float* out = (float*)d_out;                   // [8192, 32]
  float* ysc = (float*)d_ws;                    // 1 MB intermediate

  dim3 grid(N_DIM / 16); // 512 strips
  dim3 block(256);       // 8 wave32

  gemm_ut_x_scale<<<grid, block, 0, stream>>>(U, x, filters, ysc);
  gemm_u_y<<<grid, block, 0, stream>>>(U, ysc, out);
}